// Eikonal2D_27298812133648
// MI455X (gfx1250) — compile-verified
//
#include <hip/hip_runtime.h>
#include <stdint.h>

// Eikonal |grad u| = f on 512x512 (H=DX=DY=1 so fh == f), Godunov-upwind
// Jacobi x600, then quirk-faithful bilinear gather at 4M points.
//
// MI455X strategy: 3MB state is fully L2-resident, so the solve is launch-
// latency bound. Fuse 8 exact Jacobi steps per launch on 48x48 LDS tiles
// (8-wide halo, shrinking valid region => written 32x32 center is bit-exact
// 8-step Jacobi). 75 launches total. Interior tiles are staged with ONE
// Tensor-Data-Mover DMA per tile (tensor_load_to_lds, TENSORcnt); boundary
// tiles use per-lane async global->LDS copies (ASYNCcnt) with INF halo fill.

#define GM 512
#define GN 512
#define U_INF 1e5f
#define TILE 32
#define HALO 8
#define SPAN (TILE + 2 * HALO)   // 48
#define SPAN2 (SPAN * SPAN)      // 2304
#define NT 256
#define OUTER 75                 // 75 * 8 = 600 Jacobi iterations

typedef unsigned int u32;
typedef unsigned long long u64;

#if defined(__gfx1250__)
typedef u32 u32x4 __attribute__((ext_vector_type(4)));
typedef u32 u32x8 __attribute__((ext_vector_type(8)));

// Per-lane async DMA: LDS byte address (low 32 bits of flat LDS pointer) in a
// VGPR, 64-bit global address in a VGPR pair. Tracked by ASYNCcnt.
#define ASYNC_LDS_LOAD_B32(ldsaddr, gaddr)                                     \
  asm volatile("global_load_async_to_lds_b32 %0, %1, off" ::"v"(ldsaddr),      \
               "v"(gaddr)                                                      \
               : "memory")
#define WAIT_ASYNC() asm volatile("s_wait_asynccnt 0" ::: "memory")
#define WAIT_TENSOR() asm volatile("s_wait_tensorcnt 0" ::: "memory")

// One TDM descriptor: 512x512 f32 tensor, 48x48 tile, row pitch 512 elements.
// D# layout per CDNA5 ISA 8.3/8.4 (group0: count|lds_addr|global_addr|type,
// group1: data_size|tensor_dims|tile_dims|strides).
__device__ __forceinline__ void tdm_load_tile_2d(u32 lds_addr, u64 gaddr) {
  u32x4 g0;
  g0[0] = 1u;                                              // count = 1
  g0[1] = lds_addr;                                        // LDS byte address
  g0[2] = (u32)gaddr;                                      // global_addr[31:0]
  g0[3] = ((u32)(gaddr >> 32) & 0x01FFFFFFu) | (2u << 30); // [56:32] | type=2
  u32x8 g1;
  g1[0] = 2u << 16;                  // workgroup_mask=0, data_size=2 (4 bytes)
  g1[1] = (u32)GN << 16;             // tensor_dim0[15:0] = 512
  g1[2] = (u32)GM << 16;             // tensor_dim0[31:16]=0, tensor_dim1[15:0]
  g1[3] = (u32)SPAN << 16;           // tensor_dim1[31:16]=0, tile_dim0 = 48
  g1[4] = (u32)SPAN;                 // tile_dim1 = 48, tile_dim2 = 0 (2D)
  g1[5] = (u32)GN;                   // tensor_dim0_stride[31:0] = 512
  g1[6] = 0u;                        // stride0[47:32]=0, stride1[15:0]=0
  g1[7] = (u32)((GM * GN) >> 16);    // stride1[47:16] (512*512 = 0x40000)
  asm volatile("tensor_load_to_lds %0, %1" ::"s"(g0), "s"(g1) : "memory");
}
#endif

__global__ __launch_bounds__(NT) void eik_init(float* __restrict__ u,
                                               const int* __restrict__ m1,
                                               const int* __restrict__ n1) {
  const int i = blockIdx.x * NT + threadIdx.x;
  if (i < GM * GN) {
    u[i] = (i == (*m1) * GN + (*n1)) ? 0.0f : U_INF;
  }
}

// Per-lane staging path (boundary tiles on gfx1250; all tiles elsewhere).
__device__ __forceinline__ void stage_per_lane(const float* __restrict__ uin,
                                               const float* __restrict__ f,
                                               float* sU0, float* sF, int r0,
                                               int c0, int tid) {
  for (int i = tid; i < SPAN2; i += NT) {
    const int lr = i / SPAN;
    const int lc = i - lr * SPAN;
    const int gr = r0 + lr;
    const int gc = c0 + lc;
    if (gr >= 0 && gr < GM && gc >= 0 && gc < GN) {
      const int g = gr * GN + gc;
#if defined(__gfx1250__)
      ASYNC_LDS_LOAD_B32((u32)(uintptr_t)&sU0[i], (u64)(uintptr_t)(uin + g));
      ASYNC_LDS_LOAD_B32((u32)(uintptr_t)&sF[i], (u64)(uintptr_t)(f + g));
#else
      sU0[i] = uin[g];
      sF[i] = f[g];
#endif
    } else {
      sU0[i] = U_INF;  // off-grid halo stays "infinite"
      sF[i] = 1.0f;    // never read (update is guarded to in-grid cells)
    }
  }
}

__global__ __launch_bounds__(NT) void eik_step(const float* __restrict__ uin,
                                               const float* __restrict__ f,
                                               float* __restrict__ uout) {
  __shared__ float sU[2][SPAN2];  // Jacobi double buffer
  __shared__ float sF[SPAN2];     // slowness tile

  const int tid = threadIdx.x;
  const int r0 = (int)blockIdx.y * TILE - HALO;
  const int c0 = (int)blockIdx.x * TILE - HALO;

  // ---- Stage 48x48 u-tile and f-tile into LDS ----
#if defined(__gfx1250__)
  const bool interior =
      (r0 >= 0) && (c0 >= 0) && (r0 + SPAN <= GM) && (c0 + SPAN <= GN);
  if (interior) {
    // Whole tile in one TDM DMA each, issued by wave 0 only. readfirstlane
    // makes the predicate scalar-uniform -> real s_cbranch (TDM ignores EXEC).
    const unsigned lead = (unsigned)__builtin_amdgcn_readfirstlane((int)tid);
    if (lead < 32u) {
      tdm_load_tile_2d((u32)(uintptr_t)&sU[0][0],
                       (u64)(uintptr_t)(uin + r0 * GN + c0));
      tdm_load_tile_2d((u32)(uintptr_t)&sF[0],
                       (u64)(uintptr_t)(f + r0 * GN + c0));
      WAIT_TENSOR();
    }
  } else {
    stage_per_lane(uin, f, &sU[0][0], sF, r0, c0, tid);
    WAIT_ASYNC();
  }
#else
  stage_per_lane(uin, f, &sU[0][0], sF, r0, c0, tid);
#endif
  __syncthreads();

  // ---- 8 exact Jacobi steps; iteration `it` updates ring [it, SPAN-it) ----
  int cb = 0;
  for (int it = 1; it <= HALO; ++it) {
    const int lo = it, hi = SPAN - it;
    const int nb = cb ^ 1;
    for (int i = tid; i < SPAN2; i += NT) {
      const int lr = i / SPAN;
      const int lc = i - lr * SPAN;
      float v = sU[cb][i];
      if (lr >= lo && lr < hi && lc >= lo && lc < hi) {
        const int gr = r0 + lr;
        const int gc = c0 + lc;
        if (gr >= 0 && gr < GM && gc >= 0 && gc < GN) {
          const float fh = sF[i];
          const float a = fminf(sU[cb][i - SPAN], sU[cb][i + SPAN]);  // x-nbrs
          const float b = fminf(sU[cb][i - 1], sU[cb][i + 1]);        // y-nbrs
          const float amb = a - b;
          const float one_sided = fminf(a, b) + fh;
          const float disc = fmaxf(2.0f * fh * fh - amb * amb, 0.0f);
          const float two_sided = 0.5f * (a + b + sqrtf(disc));
          const float un = (fabsf(amb) >= fh) ? one_sided : two_sided;
          v = fminf(v, un);  // min-update; source cell (u==0) stays 0
        }
      }
      sU[nb][i] = v;
    }
    cb = nb;
    __syncthreads();
  }

  // ---- Write exact 32x32 center back: one b128 store per thread ----
  for (int i = tid; i < (TILE * TILE) / 4; i += NT) {
    const int lr = HALO + (i >> 3);          // 8 float4s per row
    const int lc = HALO + ((i & 7) << 2);
    const float4 v = *(const float4*)&sU[cb][lr * SPAN + lc];
    *(float4*)&uout[(r0 + lr) * GN + (c0 + lc)] = v;
  }
}

// Quirk-faithful bilinear: f_x2y1 feeds BOTH w12 and w21, f_x1y2 unused,
// 1/(DX*DY) == 1.
__device__ __forceinline__ float bilin_one(const float* __restrict__ u,
                                           float x, float y) {
  const float xu = ceilf(x), xl = floorf(x);
  const float yu = ceilf(y), yl = floorf(y);
  const int iu = (int)xu, il = (int)xl, ju = (int)yu, jl = (int)yl;
  const float f22 = u[iu * GN + ju];  // f_x2y2
  const float f21 = u[iu * GN + jl];  // f_x2y1 (used twice, per reference)
  const float f11 = u[il * GN + jl];  // f_x1y1
  const float xh = xu - x, xw = x - xl;
  const float yh = yu - y, yw = y - yl;
  return xh * yh * f11 + (xh * yw + xw * yh) * f21 + xw * yw * f22;
}

__global__ __launch_bounds__(NT) void eik_bilinear(
    const float* __restrict__ u, const float* __restrict__ gx,
    const float* __restrict__ gy, float* __restrict__ out, int q) {
  const int t = blockIdx.x * NT + threadIdx.x;
  const int base = t * 4;
  if (base + 3 < q) {
    __builtin_prefetch(gx + base + 16384, 0, 0);
    __builtin_prefetch(gy + base + 16384, 0, 0);
    const float4 x4 = *(const float4*)(gx + base);
    const float4 y4 = *(const float4*)(gy + base);
    float4 r;
    r.x = bilin_one(u, x4.x, y4.x);
    r.y = bilin_one(u, x4.y, y4.y);
    r.z = bilin_one(u, x4.z, y4.z);
    r.w = bilin_one(u, x4.w, y4.w);
    *(float4*)(out + base) = r;
  } else {
    for (int i = base; i < q; ++i) {
      out[i] = bilin_one(u, gx[i], gy[i]);
    }
  }
}

extern "C" void kernel_launch(void* const* d_in, const int* in_sizes, int n_in,
                              void* d_out, int out_size, void* d_ws,
                              size_t ws_size, hipStream_t stream) {
  (void)n_in;
  (void)out_size;
  (void)ws_size;
  const float* f = (const float*)d_in[0];
  const float* gx = (const float*)d_in[1];
  const float* gy = (const float*)d_in[2];
  const int* m1 = (const int*)d_in[3];
  const int* n1 = (const int*)d_in[4];

  float* ws0 = (float*)d_ws;   // u ping (1 MB)
  float* ws1 = ws0 + GM * GN;  // u pong (1 MB)

  eik_init<<<(GM * GN + NT - 1) / NT, NT, 0, stream>>>(ws0, m1, n1);

  float* cur = ws0;
  float* nxt = ws1;
  const dim3 grid(GN / TILE, GM / TILE);  // 16 x 16 = 256 blocks
  for (int k = 0; k < OUTER; ++k) {
    eik_step<<<grid, NT, 0, stream>>>(cur, f, nxt);
    float* t = cur;
    cur = nxt;
    nxt = t;
  }

  const int q = in_sizes[1];
  const int threads_needed = (q + 3) / 4;
  eik_bilinear<<<(threads_needed + NT - 1) / NT, NT, 0, stream>>>(
      cur, gx, gy, (float*)d_out, q);
}